// GCN_Encoder_19344532701200
// MI455X (gfx1250) — compile-verified
//
#include <hip/hip_runtime.h>

#define N_USERS 50000
#define N_ITEMS 50000
#define NNODES  100000   // N_USERS + N_ITEMS
#define DIM     64
#define NLAYERS 3
#define N_EDGES 1250000
#define BN_EPS  1e-5f

typedef __attribute__((ext_vector_type(2))) float v2f;
typedef __attribute__((ext_vector_type(8))) float v8f;

// ---------------------------------------------------------------------------
// Degree / normalization
// ---------------------------------------------------------------------------
__global__ void deg_init(float* __restrict__ deg) {
    int i = blockIdx.x * blockDim.x + threadIdx.x;
    if (i < NNODES) deg[i] = 1.0f;               // self-loop contributes 1
}

__global__ void deg_accum(const int* __restrict__ dst, float* __restrict__ deg) {
    int e = blockIdx.x * blockDim.x + threadIdx.x;
    if (e < N_EDGES) atomicAdd(&deg[dst[e]], 1.0f);
}

__global__ void deg_rsqrt(float* __restrict__ deg) {
    int i = blockIdx.x * blockDim.x + threadIdx.x;
    if (i < NNODES) deg[i] = rsqrtf(deg[i]);     // deg >= 1 always (self-loop)
}

__global__ void zero_stats(float* __restrict__ sums, float* __restrict__ sqs) {
    int i = threadIdx.x;
    if (i < DIM) { sums[i] = 0.0f; sqs[i] = 0.0f; }
}

// ---------------------------------------------------------------------------
// H = X @ W^T  via V_WMMA_F32_16X16X4_F32 (pure fp32 tensor path), with the
// bias + self-loop epilogue fused into the store:
//     H[r][c]   = h                       (read by edge scatter)
//     OUT[r][c] = bias[c] + h * dis[r]^2  (scatter accumulator init)
// One wave computes one 16x16 tile; 4 waves/block cover N-tiles 0..3 of the
// same 16-row slab.  grid.x = NNODES/16 = 6250.
// ---------------------------------------------------------------------------
__global__ __launch_bounds__(128) void gemm_xWT_fused(const float* __restrict__ X,
                                                      const float* __restrict__ W,
                                                      const float* __restrict__ bias,
                                                      const float* __restrict__ dis,
                                                      float* __restrict__ H,
                                                      float* __restrict__ OUT) {
    const int lane = threadIdx.x & 31;
    const int wave = threadIdx.x >> 5;       // n-tile 0..3
    const int m0   = blockIdx.x * 16;
    const int n0   = wave * 16;
    const int r    = lane & 15;              // A: row-in-tile, B: col-in-tile
    const int kh   = (lane >> 4) * 2;        // lanes 16..31 hold K+2,K+3

    const float* __restrict__ xrow = X + (size_t)(m0 + r) * DIM;
    const float* __restrict__ wrow = W + (size_t)(n0 + r) * DIM;  // B[k][n]=W[n][k]

    v8f c = {};
#pragma unroll
    for (int kt = 0; kt < DIM / 4; ++kt) {
        const int k0 = kt * 4 + kh;          // even -> 8B aligned float2 loads
        v2f a = *(const v2f*)(xrow + k0);
        v2f b = *(const v2f*)(wrow + k0);
        c = __builtin_amdgcn_wmma_f32_16x16x4_f32(
                /*neg_a=*/false, a, /*neg_b=*/false, b,
                /*c_mod=*/(short)0, c, /*reuse_a=*/false, /*reuse_b=*/false);
    }

    // D layout: VGPR i -> row (m0 + i + 8*(lane>=16)), col n0 + lane%16
    const int   col     = n0 + (lane & 15);
    const int   rowbase = m0 + ((lane >> 4) << 3);
    const float bcol    = bias[col];
#pragma unroll
    for (int i = 0; i < 8; ++i) {
        const size_t off = (size_t)(rowbase + i) * DIM + col;
        const float  h   = c[i];
        const float  d   = dis[rowbase + i];
        H[off]   = h;
        OUT[off] = bcol + h * (d * d);
    }
}

// ---------------------------------------------------------------------------
// Edge scatter: OUT[dst] += H[src] * dis[src]*dis[dst]
// 16 lanes per edge, float4 gather + 4 fp32 atomics each. L2-resident.
// ---------------------------------------------------------------------------
__global__ __launch_bounds__(256) void scatter_edges(const float4* __restrict__ H4,
                                                     const int*    __restrict__ src,
                                                     const int*    __restrict__ dst,
                                                     const float*  __restrict__ dis,
                                                     float* __restrict__ OUT) {
    int t = blockIdx.x * blockDim.x + threadIdx.x;
    int e = t >> 4;
    if (e >= N_EDGES) return;
    int g = t & 15;                          // feature group: floats 4g..4g+3
    int s = src[e], d = dst[e];
    float nrm = dis[s] * dis[d];
    float4 h = H4[(size_t)s * 16 + g];
    float* o = OUT + (size_t)d * DIM + g * 4;
    atomicAdd(o + 0, h.x * nrm);
    atomicAdd(o + 1, h.y * nrm);
    atomicAdd(o + 2, h.z * nrm);
    atomicAdd(o + 3, h.w * nrm);
}

// ---------------------------------------------------------------------------
// BatchNorm: per-feature sum / sumsq, LDS tree + 64 global atomics per block
// ---------------------------------------------------------------------------
__global__ __launch_bounds__(256) void bn_stats(const float* __restrict__ X,
                                                float* __restrict__ sums,
                                                float* __restrict__ sqs) {
    const int f  = threadIdx.x & 63;
    const int rg = threadIdx.x >> 6;         // 0..3 row groups
    float s = 0.0f, q = 0.0f;
    for (int r = blockIdx.x * 4 + rg; r < NNODES; r += gridDim.x * 4) {
        float v = X[(size_t)r * DIM + f];
        s += v;
        q += v * v;
    }
    __shared__ float sh[256], shq[256];
    sh[threadIdx.x]  = s;
    shq[threadIdx.x] = q;
    __syncthreads();
    if (rg == 0) {
        s = sh[f]  + sh[64 + f]  + sh[128 + f]  + sh[192 + f];
        q = shq[f] + shq[64 + f] + shq[128 + f] + shq[192 + f];
        atomicAdd(&sums[f], s);
        atomicAdd(&sqs[f],  q);
    }
}

__global__ void bn_finalize(const float* __restrict__ sums,
                            const float* __restrict__ sqs,
                            const float* __restrict__ gamma,
                            const float* __restrict__ beta,
                            float* __restrict__ scale,
                            float* __restrict__ shift) {
    int f = threadIdx.x;
    if (f >= DIM) return;
    float mean = sums[f] * (1.0f / NNODES);
    float var  = sqs[f] * (1.0f / NNODES) - mean * mean;
    float sc   = gamma[f] * rsqrtf(var + BN_EPS);
    scale[f] = sc;
    shift[f] = beta[f] - mean * sc;
}

__global__ void bn_apply(const float* __restrict__ X,
                         const float* __restrict__ scale,
                         const float* __restrict__ shift,
                         float* __restrict__ OUT) {
    int idx = blockIdx.x * blockDim.x + threadIdx.x;
    if (idx >= NNODES * DIM) return;
    int f = idx & 63;
    OUT[idx] = X[idx] * scale[f] + shift[f];
}

// ---------------------------------------------------------------------------
// Launcher
// ---------------------------------------------------------------------------
extern "C" void kernel_launch(void* const* d_in, const int* in_sizes, int n_in,
                              void* d_out, int out_size, void* d_ws, size_t ws_size,
                              hipStream_t stream) {
    (void)in_sizes; (void)n_in; (void)out_size; (void)ws_size;

    const float* x     = (const float*)d_in[0];
    const int*   ei    = (const int*)  d_in[1];   // (2, N_EDGES)
    const float* Ws    = (const float*)d_in[2];   // (3, 64, 64)
    const float* bs    = (const float*)d_in[3];   // (3, 64)
    const float* gamma = (const float*)d_in[4];
    const float* beta  = (const float*)d_in[5];
    const int* src = ei;
    const int* dst = ei + N_EDGES;

    const size_t featBytes = (size_t)NNODES * DIM * sizeof(float);
    char* ws = (char*)d_ws;
    float* bufA = (float*)ws; ws += featBytes;
    float* bufB = (float*)ws; ws += featBytes;
    float* bufH = (float*)ws; ws += featBytes;
    float* dis  = (float*)ws; ws += ((size_t)NNODES * sizeof(float) + 255) & ~(size_t)255;
    float* sums  = (float*)ws; ws += 256;
    float* sqs   = (float*)ws; ws += 256;
    float* scale = (float*)ws; ws += 256;
    float* shift = (float*)ws; ws += 256;

    // deg_inv_sqrt (self-loop included, so deg >= 1 and rsqrt is safe)
    deg_init <<<(NNODES + 255) / 256, 256, 0, stream>>>(dis);
    deg_accum<<<(N_EDGES + 255) / 256, 256, 0, stream>>>(dst, dis);
    deg_rsqrt<<<(NNODES + 255) / 256, 256, 0, stream>>>(dis);
    zero_stats<<<1, 128, 0, stream>>>(sums, sqs);

    const float* cur = x;
    float* nxt   = bufA;
    float* other = bufB;
    for (int L = 0; L < NLAYERS; ++L) {
        const float* W = Ws + (size_t)L * DIM * DIM;
        const float* b = bs + (size_t)L * DIM;
        // GEMM + fused bias/self-loop epilogue: writes H and initializes nxt
        gemm_xWT_fused<<<NNODES / 16, 128, 0, stream>>>(cur, W, b, dis, bufH, nxt);
        scatter_edges<<<((size_t)N_EDGES * 16 + 255) / 256, 256, 0, stream>>>(
            (const float4*)bufH, src, dst, dis, nxt);
        cur = nxt;
        float* t = nxt; nxt = other; other = t;
    }

    bn_stats<<<256, 256, 0, stream>>>(cur, sums, sqs);
    bn_finalize<<<1, 64, 0, stream>>>(sums, sqs, gamma, beta, scale, shift);
    bn_apply<<<(NNODES * DIM + 255) / 256, 256, 0, stream>>>(cur, scale, shift, (float*)d_out);
}